// Mapper_10462540333249
// MI455X (gfx1250) — compile-verified
//
#include <hip/hip_runtime.h>

// Problem constants (fixed by the reference)
#define D_ROW   128
#define D_COL   512
#define D_ALL   (D_ROW * D_COL)   // 65536 elements per x row / per table
#define BATCH   1024

// Native ext-vector types (required by the amdgcn builtins)
typedef int   v4i __attribute__((ext_vector_type(4)));
typedef float v4f __attribute__((ext_vector_type(4)));

// Explicit LDS (address_space(3)) pointer types for guaranteed ds_* lowering
typedef __attribute__((address_space(3))) void  as3_void;
typedef __attribute__((address_space(3))) char  lds_char;
typedef __attribute__((address_space(3))) float lds_float;

// ---------------------------------------------------------------------------
// CDNA5 async global->LDS availability probe
// ---------------------------------------------------------------------------
#if defined(__HIP_DEVICE_COMPILE__) && defined(__gfx1250__)
#  if __has_builtin(__builtin_amdgcn_global_load_async_to_lds_b128)
#    define ATHENA_ASYNC_LDS 1
#  endif
#endif

// ---------------------------------------------------------------------------
// Stage 2 (defined first so it appears in the disasm snippet):
// one workgroup per batch row.
//   - async-copy the full 256KB x row into LDS (fits CDNA5's 320KB/WGP)
//   - each of 1024 threads emits 16 coalesced float4 outputs:
//       out[b, g] = LDS_x[ind_t[g]] * w_t[g],   g = m*512 + r
//     (ind_t holds BYTE offsets, pre-scaled in prep_kernel)
// ---------------------------------------------------------------------------
__global__ __launch_bounds__(1024) void map_kernel(
    const float* __restrict__ x,
    const int*   __restrict__ ind_t,   // byte offsets into the LDS row
    const float* __restrict__ w_t,
    float*       __restrict__ out)
{
    __shared__ float smem[D_ALL];             // 256 KB LDS
    const int b   = blockIdx.x;
    const int tid = threadIdx.x;              // 0..1023

    const v4f* __restrict__ xrow4 = (const v4f*)(x + (size_t)b * D_ALL);
    v4f* smem4 = (v4f*)smem;

#if defined(ATHENA_ASYNC_LDS)
    {
        #pragma unroll
        for (int i = 0; i < 16; ++i) {
            int e4 = tid + i * 1024;          // float4 index, 0..16383
            __builtin_amdgcn_global_load_async_to_lds_b128(
                (v4i*)(xrow4 + e4),           // global source (16B per lane)
                (v4i*)(smem4 + e4),           // LDS destination
                0 /*offset*/, 0 /*cpol*/);
        }
#  if __has_builtin(__builtin_amdgcn_s_wait_asynccnt)
        __builtin_amdgcn_s_wait_asynccnt(0);
#  else
        asm volatile("s_wait_asynccnt 0" ::: "memory");
#  endif
    }
#else
    {
        #pragma unroll
        for (int i = 0; i < 16; ++i) {
            int e4 = tid + i * 1024;
            smem4[e4] = xrow4[e4];
        }
    }
#endif
    __syncthreads();

    const lds_char* sb = (const lds_char*)((as3_void*)smem);

    const v4i* __restrict__ it4 = (const v4i*)ind_t;
    const v4f* __restrict__ wt4 = (const v4f*)w_t;
    v4f*       __restrict__ o4  = (v4f*)(out + (size_t)b * D_ALL);

    #pragma unroll 4
    for (int k = 0; k < 16; ++k) {
        int g4 = tid + k * 1024;              // float4 index into the 65536 outs
        v4i off = it4[g4];                    // byte offsets; coalesced, L2-hot
        v4f wv  = wt4[g4];                    // coalesced, L2-hot
        v4f o;
        o.x = *(const lds_float*)(sb + off.x) * wv.x;   // ds_load_b32, no shift
        o.y = *(const lds_float*)(sb + off.y) * wv.y;
        o.z = *(const lds_float*)(sb + off.z) * wv.z;
        o.w = *(const lds_float*)(sb + off.w) * wv.w;
        __builtin_nontemporal_store(o, o4 + g4);        // streaming b128 store
    }
}

// ---------------------------------------------------------------------------
// Stage 1: build transposed gather tables in workspace.
//   ind_t[m*512 + r] = 4 * ind[r*128 + m]            (byte offset)
//   w_t  [m*512 + r] = filters[r*65536 + ind[r*128 + m]]
// ---------------------------------------------------------------------------
__global__ __launch_bounds__(256) void prep_kernel(
    const float* __restrict__ filters,
    const int*   __restrict__ ind,
    int*         __restrict__ ind_t,
    float*       __restrict__ w_t)
{
    int g = blockIdx.x * 256 + threadIdx.x;   // g = m*512 + r, 0..65535
    int m = g >> 9;
    int r = g & (D_COL - 1);
    int j = ind[r * D_ROW + m];
    ind_t[g] = j << 2;                        // pre-scaled to bytes
    w_t[g]   = filters[(size_t)r * D_ALL + j];
}

// ---------------------------------------------------------------------------
// Launch
// ---------------------------------------------------------------------------
extern "C" void kernel_launch(void* const* d_in, const int* in_sizes, int n_in,
                              void* d_out, int out_size, void* d_ws, size_t ws_size,
                              hipStream_t stream) {
    const float* x       = (const float*)d_in[0];  // (1024, 65536) f32
    const float* filters = (const float*)d_in[1];  // (512, 65536)  f32
    const int*   ind     = (const int*)  d_in[2];  // (512, 128)    i32
    float*       out     = (float*)d_out;          // (1024, 128, 512) f32

    int*   ind_t = (int*)d_ws;                                // 256 KB
    float* w_t   = (float*)((char*)d_ws + (size_t)D_ALL * 4); // 256 KB

    prep_kernel<<<D_ALL / 256, 256, 0, stream>>>(filters, ind, ind_t, w_t);
    map_kernel<<<BATCH, 1024, 0, stream>>>(x, ind_t, w_t, out);
}